// DRR_72138270704027
// MI455X (gfx1250) — compile-verified
//
#include <hip/hip_runtime.h>
#include <math.h>

// DRR via incremental Siddon ray marching (replaces reference's 771-element
// per-ray sort). One ray is split across 4 waves (disjoint alpha quarters),
// partials reduced in LDS. Volume (64MB) is L2-resident on MI455X (192MB L2);
// the loop processes TWO segments per iteration so each wave keeps two
// independent gathers in flight (MLP=2) on top of 8192-wave occupancy.

#define HH 256
#define WW 256
#define NRAYS (HH * WW)
#define DELX 2.0f
#define DELY 2.0f
#define EPSR 1e-8f

__global__ __launch_bounds__(256) void drr_siddon_kernel(
    const float* __restrict__ vol,      // 256^3, C-order [x][y][z]
    const float* __restrict__ spacing,  // 3
    const float* __restrict__ sdrp,     // 1
    const float* __restrict__ rot,      // 3 (theta, phi, gamma)
    const float* __restrict__ trans,    // 3
    float* __restrict__ out)            // 65536
{
    __shared__ float s_part[4 * 64];
    __shared__ float s_rayl[64];

    const int lane = threadIdx.x & 31;
    const int wv   = threadIdx.x >> 5;   // wave in block: 0..7
    const int q    = wv & 3;             // alpha quarter 0..3
    const int half = wv >> 2;            // 0..1
    const int lray = half * 32 + lane;   // local ray 0..63
    const int ray  = blockIdx.x * 64 + lray;
    const int i    = ray >> 8;           // detector row (t index)
    const int j    = ray & 255;          // detector col (s index)

    // ---- rotation R = Rz(theta) @ Ry(phi) @ Rx(gamma) (wave-uniform, SALU) ----
    const float th = rot[0], ph = rot[1], ga = rot[2];
    const float ct = cosf(th), st = sinf(th);
    const float cp = cosf(ph), sp = sinf(ph);
    const float cg = cosf(ga), sg = sinf(ga);
    const float r0x = ct * cp, r0y = st * cp, r0z = -sp;          // R[:,0]
    const float ux = -st * cg + ct * sp * sg;                     // R[:,1]
    const float uy =  ct * cg + st * sp * sg;
    const float uz =  cp * sg;
    const float vx =  st * sg + ct * sp * cg;                     // R[:,2]
    const float vy = -ct * sg + st * sp * cg;
    const float vz =  cp * cg;

    const float sdr = sdrp[0];
    const float tx = trans[0], ty = trans[1], tz = trans[2];
    const float sx = sdr * r0x + tx;          // source
    const float sy = sdr * r0y + ty;
    const float sz = sdr * r0z + tz;
    const float cex = -sdr * r0x + tx;        // detector center
    const float cey = -sdr * r0y + ty;
    const float cez = -sdr * r0z + tz;

    const float t = ((float)(i - (HH / 2)) + 1.0f) * DELX;
    const float s = ((float)(j - (WW / 2)) + 1.0f) * DELY;

    const float tgx = t * ux + s * vx + cex;
    const float tgy = t * uy + s * vy + cey;
    const float tgz = t * uz + s * vz + cez;

    // sdd = target - source + EPS (reference adds EPS per-component)
    const float dx = tgx - sx + EPSR;
    const float dy = tgy - sy + EPSR;
    const float dz = tgz - sz + EPSR;

    const float spx = spacing[0], spy = spacing[1], spz = spacing[2];
    const float ispx = 1.0f / spx, ispy = 1.0f / spy, ispz = 1.0f / spz;
    const float idx_ = 1.0f / dx, idy_ = 1.0f / dy, idz_ = 1.0f / dz;

    // entry/exit alphas over planes k = 0..256 (dims-1 == 256)
    const float a0x = (0.0f - sx) * idx_, a1x = (256.0f * spx - sx) * idx_;
    const float a0y = (0.0f - sy) * idy_, a1y = (256.0f * spy - sy) * idy_;
    const float a0z = (0.0f - sz) * idz_, a1z = (256.0f * spz - sz) * idz_;

    const float amin = fmaxf(fmaxf(fminf(a0x, a1x), fminf(a0y, a1y)), fminf(a0z, a1z));
    const float amax = fminf(fminf(fmaxf(a0x, a1x), fmaxf(a0y, a1y)), fmaxf(a0z, a1z));

    const float raylen = sqrtf(dx * dx + dy * dy + dz * dz);

    // position(alpha)/spacing as a single FMA per axis:
    //   idx(alpha) = trunc(fma(alpha, d*isp, s*isp))
    const float dxi = dx * ispx, dyi = dy * ispy, dzi = dz * ispz;
    const float sxi = sx * ispx, syi = sy * ispy, szi = sz * ispz;

    float acc = 0.0f;
    if (amax > amin) {
        const float range   = amax - amin;
        const float a_start = amin + 0.25f * (float)q * range;
        const float a_end   = (q == 3) ? amax : (amin + 0.25f * (float)(q + 1) * range);

        // position at a_start
        const float px = sx + a_start * dx;
        const float py = sy + a_start * dy;
        const float pz = sz + a_start * dz;

        // first plane crossing strictly after a_start, per axis, + alpha step
        float ax, ay, az, dax, day, daz;
        if (dx > 0.0f) { float k = floorf(px * ispx) + 1.0f; ax = (k * spx - sx) * idx_; dax =  spx * idx_; }
        else           { float k = ceilf (px * ispx) - 1.0f; ax = (k * spx - sx) * idx_; dax = -spx * idx_; }
        if (dy > 0.0f) { float k = floorf(py * ispy) + 1.0f; ay = (k * spy - sy) * idy_; day =  spy * idy_; }
        else           { float k = ceilf (py * ispy) - 1.0f; ay = (k * spy - sy) * idy_; day = -spy * idy_; }
        if (dz > 0.0f) { float k = floorf(pz * ispz) + 1.0f; az = (k * spz - sz) * idz_; daz =  spz * idz_; }
        else           { float k = ceilf (pz * ispz) - 1.0f; az = (k * spz - sz) * idz_; daz = -spz * idz_; }

        float acur = a_start;
        // Two segments per iteration: both gathers issue before either FMA,
        // so each wave keeps 2 loads in flight. 640*2 > 774 max segments.
        for (int it = 0; (it < 640) && (acur < a_end); ++it) {
            // ---- segment A: [acur, e1] ----
            const float n1   = fminf(ax, fminf(ay, az));
            const bool  adv1 = (n1 < a_end);
            const float e1   = adv1 ? n1 : a_end;

            // conditional axis advance #1 (branchless cndmask+add)
            const bool cx1 = adv1 && (ax <= n1);
            const bool cy1 = adv1 && (ay <= n1);
            const bool cz1 = adv1 && (az <= n1);
            ax = cx1 ? (ax + dax) : ax;
            ay = cy1 ? (ay + day) : ay;
            az = cz1 ? (az + daz) : az;

            // ---- segment B: [e1, e2] (empty if !adv1) ----
            const float n2   = fminf(ax, fminf(ay, az));
            const bool  adv2 = adv1 && (n2 < a_end);
            const float e2   = adv2 ? n2 : a_end;     // == a_end when !adv1 too

            const float midA  = 0.5f * (acur + e1);
            const float stepA = e1 - acur;
            const float midB  = 0.5f * (e1 + e2);
            const float stepB = adv1 ? (e2 - e1) : 0.0f;

            // voxel indices: trunc-toward-zero (v_cvt_i32_f32) + med3 clamp
            int ixA = __float2int_rz(fmaf(midA, dxi, sxi));
            int iyA = __float2int_rz(fmaf(midA, dyi, syi));
            int izA = __float2int_rz(fmaf(midA, dzi, szi));
            int ixB = __float2int_rz(fmaf(midB, dxi, sxi));
            int iyB = __float2int_rz(fmaf(midB, dyi, syi));
            int izB = __float2int_rz(fmaf(midB, dzi, szi));
            ixA = min(max(ixA, 0), 255); iyA = min(max(iyA, 0), 255); izA = min(max(izA, 0), 255);
            ixB = min(max(ixB, 0), 255); iyB = min(max(iyB, 0), 255); izB = min(max(izB, 0), 255);

            // two independent gathers (usually same/adjacent cache line)
            const float vA = vol[(((ixA << 8) | iyA) << 8) | izA];
            const float vB = vol[(((ixB << 8) | iyB) << 8) | izB];
            acc = fmaf(vA, stepA, acc);
            acc = fmaf(vB, stepB, acc);

            // conditional axis advance #2
            const bool cx2 = adv2 && (ax <= n2);
            const bool cy2 = adv2 && (ay <= n2);
            const bool cz2 = adv2 && (az <= n2);
            ax = cx2 ? (ax + dax) : ax;
            ay = cy2 ? (ay + day) : ay;
            az = cz2 ? (az + daz) : az;

            acur = e2;   // == a_end on the closing iteration -> loop exits
        }
    }

    // ---- LDS reduction: 4 quarter-partials -> one ray ----
    s_part[q * 64 + lray] = acc;
    if (q == 0) s_rayl[lray] = raylen;
    __syncthreads();

    if (threadIdx.x < 64) {
        const int tt = threadIdx.x;
        const float sum = s_part[tt] + s_part[64 + tt] + s_part[128 + tt] + s_part[192 + tt];
        out[blockIdx.x * 64 + tt] = sum * s_rayl[tt];
    }
}

extern "C" void kernel_launch(void* const* d_in, const int* in_sizes, int n_in,
                              void* d_out, int out_size, void* d_ws, size_t ws_size,
                              hipStream_t stream) {
    (void)in_sizes; (void)n_in; (void)d_ws; (void)ws_size; (void)out_size;
    const float* vol     = (const float*)d_in[0];
    const float* spacing = (const float*)d_in[1];
    const float* sdr     = (const float*)d_in[2];
    const float* rot     = (const float*)d_in[3];
    const float* trans   = (const float*)d_in[4];
    float* out = (float*)d_out;

    // 64 rays per block x 4 alpha-quarters = 256 threads (8 wave32s).
    // 1024 blocks -> 8192 waves total for gather-latency hiding.
    drr_siddon_kernel<<<NRAYS / 64, 256, 0, stream>>>(vol, spacing, sdr, rot, trans, out);
}